// GCN_43009802502548
// MI455X (gfx1250) — compile-verified
//
#include <hip/hip_runtime.h>
#include <math.h>

#define DFEAT 128

typedef __attribute__((ext_vector_type(2))) float v2f;
typedef __attribute__((ext_vector_type(8))) float v8f;

// ---------------------------------------------------------------------------
// Degree: dis[i] = 1 (self loop) + #in-edges, then dis = rsqrt(deg)
// ---------------------------------------------------------------------------
__global__ void k_deg_init(float* __restrict__ dis, int N) {
    int i = blockIdx.x * blockDim.x + threadIdx.x;
    if (i < N) dis[i] = 1.0f;
}

__global__ void k_deg_count(const int* __restrict__ dst, int E,
                            float* __restrict__ dis) {
    int e = blockIdx.x * blockDim.x + threadIdx.x;
    if (e < E) unsafeAtomicAdd(&dis[dst[e]], 1.0f);
}

__global__ void k_deg_finish(float* __restrict__ dis, int N) {
    int i = blockIdx.x * blockDim.x + threadIdx.x;
    if (i < N) dis[i] = rsqrtf(dis[i]);
}

// ---------------------------------------------------------------------------
// agg1 init with self-loop contribution: agg1[i] = x[i] * dis[i]
// one thread per float4 (32 threads per node row)
// ---------------------------------------------------------------------------
__global__ void k_init_agg1(const float* __restrict__ x,
                            const float* __restrict__ dis,
                            float* __restrict__ agg, int N) {
    int t = blockIdx.x * blockDim.x + threadIdx.x;
    int i = t >> 5, c = t & 31;
    if (i < N) {
        float4 v = ((const float4*)x)[i * 32 + c];
        float s = dis[i];
        v.x *= s; v.y *= s; v.z *= s; v.w *= s;
        ((float4*)agg)[i * 32 + c] = v;
    }
}

// ---------------------------------------------------------------------------
// Edge scatter: agg[dst[e]] += hsrc[src[e]] * (apply_scale ? dis[src[e]] : 1)
// One wave per edge, one float4 per lane -> 4 hw fp32 atomics per lane.
// Index stream prefetched ahead (global_prefetch_b8).
// ---------------------------------------------------------------------------
__global__ void k_scatter(const float* __restrict__ hsrc,
                          const float* __restrict__ dis,
                          const int* __restrict__ src,
                          const int* __restrict__ dst,
                          int E, float* __restrict__ agg, int apply_scale) {
    int e = blockIdx.x * 8 + (threadIdx.x >> 5);
    if (e >= E) return;
    int c = threadIdx.x & 31;
    if (c == 0) {
        // keep the edge-index stream ~4K edges ahead of the gather/atomics
        int pe = e + 4096;
        if (pe < E) {
            __builtin_prefetch(&src[pe], 0, 1);
            __builtin_prefetch(&dst[pe], 0, 1);
        }
    }
    int s = src[e], d = dst[e];
    float sc = apply_scale ? dis[s] : 1.0f;
    float4 v = ((const float4*)hsrc)[(size_t)s * 32 + c];
    float* p = agg + (size_t)d * DFEAT + c * 4;
    unsafeAtomicAdd(p + 0, v.x * sc);
    unsafeAtomicAdd(p + 1, v.y * sc);
    unsafeAtomicAdd(p + 2, v.z * sc);
    unsafeAtomicAdd(p + 3, v.w * sc);
}

// ---------------------------------------------------------------------------
// Layer-1 GEMM (fp32 WMMA 16x16x4):
//   h1s[row] = dis[row] * relu( (dis[row]*agg1[row]) @ W1 + b1 )
// written to BOTH outB (gather source for layer 2) and outA (agg2 self-loop
// init) in the epilogue. 256 threads = 8 wave32; each wave computes a 16x128
// output stripe; W1 staged in LDS; 256 v_wmma per wave. Interior waves take
// an unguarded (EXEC-uniform) store path; only the tail wave predicates.
// ---------------------------------------------------------------------------
__global__ __launch_bounds__(256)
void k_gemm1_wmma(const float* __restrict__ agg,
                  const float* __restrict__ W,
                  const float* __restrict__ bias,
                  const float* __restrict__ dis,
                  float* __restrict__ outA,
                  float* __restrict__ outB, int N) {
    __shared__ float sW[DFEAT * DFEAT];
    __shared__ float sB[DFEAT];
    int tid = threadIdx.x;
    for (int i = tid; i < DFEAT * DFEAT; i += 256) sW[i] = W[i];
    if (tid < DFEAT) sB[tid] = bias[tid];
    __syncthreads();

    int wave  = tid >> 5;
    int lane  = tid & 31;
    int m     = lane & 15;       // row-in-tile (A) / col-in-tile (B)
    int khalf = lane >> 4;       // K-pair selector
    int rb    = blockIdx.x * 128 + wave * 16;

    // A-fragment row (clamped for OOB tail; stores are predicated instead)
    int rowA = rb + m;
    if (rowA > N - 1) rowA = N - 1;
    float scaleA = dis[rowA];                   // dst-side norm folded in
    const float* Arow = agg + (size_t)rowA * DFEAT;

    v8f acc[8] = {};                            // 8 column tiles of 16x16 f32

    for (int k0 = 0; k0 < DFEAT; k0 += 4) {
        int kk = k0 + 2 * khalf;                // A/B: VGPR0->kk, VGPR1->kk+1
        v2f a;
        a.x = Arow[kk]     * scaleA;
        a.y = Arow[kk + 1] * scaleA;
#pragma unroll
        for (int nt = 0; nt < 8; ++nt) {
            int col = nt * 16 + m;
            v2f b;
            b.x = sW[kk * DFEAT + col];
            b.y = sW[(kk + 1) * DFEAT + col];
            acc[nt] = __builtin_amdgcn_wmma_f32_16x16x4_f32(
                false, a, false, b, (short)0, acc[nt], false, false);
        }
    }

    // Epilogue: bias + relu + src-side norm for layer 2; dual store.
    if (rb + 16 <= N) {
        // wave-uniform interior fast path: no per-element predication
#pragma unroll
        for (int nt = 0; nt < 8; ++nt) {
            int col = nt * 16 + m;
            float bv = sB[col];
#pragma unroll
            for (int r = 0; r < 8; ++r) {
                int row = rb + r + 8 * khalf;   // D layout: M = r + 8*khalf
                float v = fmaxf(acc[nt][r] + bv, 0.0f) * dis[row];
                size_t off = (size_t)row * DFEAT + col;
                outB[off] = v;
                outA[off] = v;
            }
        }
    } else {
#pragma unroll
        for (int nt = 0; nt < 8; ++nt) {
            int col = nt * 16 + m;
            float bv = sB[col];
#pragma unroll
            for (int r = 0; r < 8; ++r) {
                int row = rb + r + 8 * khalf;
                if (row < N) {
                    float v = fmaxf(acc[nt][r] + bv, 0.0f) * dis[row];
                    size_t off = (size_t)row * DFEAT + col;
                    outB[off] = v;
                    outA[off] = v;
                }
            }
        }
    }
}

// ---------------------------------------------------------------------------
// Weff = W2 @ Wr (128-vec), ceff = b2.Wr + br   (tiny, one block)
// ---------------------------------------------------------------------------
__global__ void k_weff(const float* __restrict__ W2,
                       const float* __restrict__ Wr,
                       const float* __restrict__ b2,
                       const float* __restrict__ br,
                       float* __restrict__ weff) {
    int k = threadIdx.x;
    if (k < DFEAT) {
        float s = 0.0f;
        for (int j = 0; j < DFEAT; ++j) s += W2[k * DFEAT + j] * Wr[j];
        weff[k] = s;
    }
    if (k == 0) {
        float c = br[0];
        for (int j = 0; j < DFEAT; ++j) c += b2[j] * Wr[j];
        weff[DFEAT] = c;
    }
}

// ---------------------------------------------------------------------------
// Fused layer-2 GEMM + regressor: out[i] = dis[i] * (agg2[i] . Weff) + ceff
// One wave per node, float4 per lane, xor-shuffle reduction (wave32).
// ---------------------------------------------------------------------------
__global__ void k_final(const float* __restrict__ agg2,
                        const float* __restrict__ dis,
                        const float* __restrict__ weff,
                        float* __restrict__ out, int N) {
    int lane = threadIdx.x & 31;
    int i = blockIdx.x * 8 + (threadIdx.x >> 5);
    if (i >= N) return;
    float4 v = ((const float4*)agg2)[(size_t)i * 32 + lane];
    float4 w = ((const float4*)weff)[lane];
    float p = v.x * w.x + v.y * w.y + v.z * w.z + v.w * w.w;
#pragma unroll
    for (int off = 16; off > 0; off >>= 1) p += __shfl_xor(p, off, 32);
    if (lane == 0) out[i] = dis[i] * p + weff[DFEAT];
}

// ---------------------------------------------------------------------------
extern "C" void kernel_launch(void* const* d_in, const int* in_sizes, int n_in,
                              void* d_out, int out_size, void* d_ws, size_t ws_size,
                              hipStream_t stream) {
    (void)n_in; (void)out_size; (void)ws_size;
    const float* x  = (const float*)d_in[0];
    const int*   src= (const int*)  d_in[1];
    const int*   dst= (const int*)  d_in[2];
    const float* W1 = (const float*)d_in[3];
    const float* b1 = (const float*)d_in[4];
    const float* W2 = (const float*)d_in[5];
    const float* b2 = (const float*)d_in[6];
    const float* Wr = (const float*)d_in[7];
    const float* br = (const float*)d_in[8];
    float* out = (float*)d_out;

    int N = in_sizes[0] / DFEAT;
    int E = in_sizes[1];

    // Workspace carve-up (256B aligned): dis | bufA(agg1/agg2) | bufB(h1s) | weff
    char* ws = (char*)d_ws;
    size_t o = 0;
    auto carve = [&](size_t bytes) -> void* {
        void* p = ws + o;
        o += (bytes + 255) & ~(size_t)255;
        return p;
    };
    float* dis  = (float*)carve((size_t)N * 4);
    float* bufA = (float*)carve((size_t)N * DFEAT * 4);
    float* bufB = (float*)carve((size_t)N * DFEAT * 4);
    float* weff = (float*)carve((DFEAT + 1) * 4);

    const int tpb   = 256;
    int gN    = (N + tpb - 1) / tpb;
    int gE    = (E + tpb - 1) / tpb;
    int gRowF4= (N * 32 + tpb - 1) / tpb;   // thread per float4 of node rows
    int gScat = (E + 7) / 8;                // wave per edge
    int gGemm = (N + 127) / 128;            // 128 rows per block
    int gFin  = (N + 7) / 8;                // wave per node

    // Degrees -> dis = rsqrt(deg)
    k_deg_init  <<<gN, tpb, 0, stream>>>(dis, N);
    k_deg_count <<<gE, tpb, 0, stream>>>(dst, E, dis);
    k_deg_finish<<<gN, tpb, 0, stream>>>(dis, N);

    // Regressor fold (independent; tiny)
    k_weff<<<1, DFEAT, 0, stream>>>(W2, Wr, b2, br, weff);

    // Layer 1: agg1 = selfloop(x*dis) + scatter(x*dis)
    k_init_agg1<<<gRowF4, tpb, 0, stream>>>(x, dis, bufA, N);
    k_scatter  <<<gScat, tpb, 0, stream>>>(x, dis, src, dst, E, bufA, /*scale=*/1);

    // Layer-1 GEMM (WMMA) -> h1s into bufB, and agg2 self-loop init into bufA
    k_gemm1_wmma<<<gGemm, tpb, 0, stream>>>(bufA, W1, b1, dis, bufA, bufB, N);

    // Layer 2 scatter: agg2 += h1s[src] (already src-normalized)
    k_scatter<<<gScat, tpb, 0, stream>>>(bufB, dis, src, dst, E, bufA, /*scale=*/0);

    // Fused layer-2 GEMM + regressor
    k_final<<<gFin, tpb, 0, stream>>>(bufA, dis, weff, out, N);
}